// ModelNew_3556232921972
// MI455X (gfx1250) — compile-verified
//
#include <hip/hip_runtime.h>
#include <math.h>

typedef __attribute__((ext_vector_type(2))) float v2f;
typedef __attribute__((ext_vector_type(8))) float v8f;

#define NB   64      // batch
#define CIN  64
#define COUT 128
#define HW   (128*128)
#define INV_AREA (1.0f/(130.0f*130.0f))

// ---------------------------------------------------------------------------
// Kernel 1 (bandwidth-dominant): S[b,ci] = sum over 128x128 spatial plane.
// 4096 planes, one block (256 thr = 8 waves) per 64KB plane, float4 loads.
// ---------------------------------------------------------------------------
__global__ void spatial_sum_kernel(const float* __restrict__ x,
                                   float* __restrict__ S) {
    __shared__ float red[256];
    const int plane = blockIdx.x;          // b*CIN + ci
    const int tid   = threadIdx.x;
    const float4* p = (const float4*)(x + (size_t)plane * HW);
    float acc = 0.0f;
    #pragma unroll
    for (int i = 0; i < 16; ++i) {          // 16384 floats / (256 thr * 4) = 16
        float4 v = p[i * 256 + tid];
        acc += (v.x + v.y) + (v.z + v.w);
    }
    red[tid] = acc;
    __syncthreads();
    for (int s = 128; s > 0; s >>= 1) {
        if (tid < s) red[tid] += red[tid + s];
        __syncthreads();
    }
    if (tid == 0) S[plane] = red[0];
}

// ---------------------------------------------------------------------------
// Kernel 2: Wsum[ci,co] = sum of the 3x3 taps. Weight layout [Cin,Cout,3,3]
// => each (ci,co) pair is 9 contiguous floats.
// ---------------------------------------------------------------------------
__global__ void wsum_kernel(const float* __restrict__ w,
                            float* __restrict__ Wsum) {
    const int idx = blockIdx.x * blockDim.x + threadIdx.x;   // 0..8191
    const float* p = w + idx * 9;
    float s = 0.0f;
    #pragma unroll
    for (int i = 0; i < 9; ++i) s += p[i];
    Wsum[idx] = s;
}

// ---------------------------------------------------------------------------
// Kernel 3: G[64,128] = S[64,64] @ Wsum[64,128] via V_WMMA_F32_16X16X4_F32.
// One wave per 16x16 tile of G; grid = (4 M-tiles, 8 N-tiles); 16 K-steps.
// Fragment layouts per ISA 7.12.2:
//   A 16x4 : lanes 0-15 -> {K=k0,k0+1}, lanes 16-31 -> {K=k0+2,k0+3}, M=lane%16
//   B 4x16 : VGPR0 -> row K=k0(+2 for hi half), VGPR1 -> K=k0+1(+2), N=lane%16
//   C/D    : VGPR r -> row r (lanes 0-15) / row r+8 (lanes 16-31), N=lane%16
// ---------------------------------------------------------------------------
__global__ void gemm_wmma_kernel(const float* __restrict__ S,
                                 const float* __restrict__ Wsum,
                                 float* __restrict__ G) {
    const int lane = threadIdx.x;          // 0..31, full wave, EXEC all-1s
    const int half = lane >> 4;            // 0 or 1
    const int l16  = lane & 15;
    const int mt   = blockIdx.x;           // 0..3
    const int nt   = blockIdx.y;           // 0..7
    const int m    = mt * 16 + l16;
    const int n    = nt * 16 + l16;

    v8f acc = {};
    #pragma unroll
    for (int k0 = 0; k0 < CIN; k0 += 4) {
        const int ka = k0 + half * 2;
        v2f a, b;
        a.x = S[m * CIN + ka];
        a.y = S[m * CIN + ka + 1];
        b.x = Wsum[ka * COUT + n];
        b.y = Wsum[(ka + 1) * COUT + n];
        acc = __builtin_amdgcn_wmma_f32_16x16x4_f32(
            /*neg_a=*/false, a, /*neg_b=*/false, b,
            /*c_mod=*/(short)0, acc, /*reuse_a=*/false, /*reuse_b=*/false);
    }
    #pragma unroll
    for (int r = 0; r < 8; ++r) {
        G[(mt * 16 + r + half * 8) * COUT + n] = acc[r];
    }
}

// ---------------------------------------------------------------------------
// Kernel 4: out[b] = 10 * logsumexp_c( G[b,c]/16900 + bias[c] )
// One block (128 thr) per batch; LDS max-reduce then sum-reduce.
// ---------------------------------------------------------------------------
__global__ void lse_kernel(const float* __restrict__ G,
                           const float* __restrict__ bias,
                           float* __restrict__ out) {
    __shared__ float red[128];
    const int b = blockIdx.x;
    const int c = threadIdx.x;
    const float z = G[b * COUT + c] * INV_AREA + bias[c];
    red[c] = z;
    __syncthreads();
    for (int s = 64; s > 0; s >>= 1) {
        if (c < s) red[c] = fmaxf(red[c], red[c + s]);
        __syncthreads();
    }
    const float mx = red[0];
    __syncthreads();
    red[c] = __expf(z - mx);
    __syncthreads();
    for (int s = 64; s > 0; s >>= 1) {
        if (c < s) red[c] += red[c + s];
        __syncthreads();
    }
    if (c == 0) out[b] = 10.0f * (mx + __logf(red[0]));
}

// ---------------------------------------------------------------------------
extern "C" void kernel_launch(void* const* d_in, const int* in_sizes, int n_in,
                              void* d_out, int out_size, void* d_ws, size_t ws_size,
                              hipStream_t stream) {
    const float* x    = (const float*)d_in[0];   // [64,64,128,128]
    const float* w    = (const float*)d_in[1];   // [64,128,3,3]
    const float* bias = (const float*)d_in[2];   // [128]
    float* out = (float*)d_out;                  // [64,1]

    float* S    = (float*)d_ws;                  // 4096 floats
    float* Wsum = S + NB * CIN;                  // 8192 floats
    float* G    = Wsum + CIN * COUT;             // 8192 floats

    spatial_sum_kernel<<<NB * CIN, 256, 0, stream>>>(x, S);
    wsum_kernel<<<(CIN * COUT) / 256, 256, 0, stream>>>(w, Wsum);
    dim3 gtiles(NB / 16, COUT / 16);             // (4, 8)
    gemm_wmma_kernel<<<gtiles, 32, 0, stream>>>(S, Wsum, G);
    lse_kernel<<<NB, 128, 0, stream>>>(G, bias, out);
}